// Codec_66967130079958
// MI455X (gfx1250) — compile-verified
//
#include <hip/hip_runtime.h>
#include <hip/hip_bf16.h>
#include <math.h>

typedef __attribute__((ext_vector_type(2))) float v2f;
typedef __attribute__((ext_vector_type(8))) float v8f;

#define IMG_H 192
#define IMG_W 192
#define INT_H 186   /* 192 - 2*3 */
#define INT_W 186
#define NCH   24    /* 8 * 3 */

// Hardware tanh on CDNA5 (V_TANH_F32) when the builtin exists.
#if __has_builtin(__builtin_amdgcn_tanhf)
#define TANH(x) __builtin_amdgcn_tanhf(x)
#elif __has_builtin(__builtin_amdgcn_tanh_f32)
#define TANH(x) __builtin_amdgcn_tanh_f32(x)
#else
#define TANH(x) tanhf(x)
#endif

__device__ __forceinline__ int iclamp(int v, int lo, int hi) {
    return v < lo ? lo : (v > hi ? hi : v);
}

// ---------------------------------------------------------------------------
// One wave32 workgroup per (b,c) channel. Diagonal wavefront: within a group
// of 16 interior rows, at step t row m computes column j = t - 4m.  The 16
// in-flight pixels form the M=16 batch of V_WMMA_F32_16X16X4_F32 chains.
// Inactive rows carry garbage (rows of D depend only on rows of A), so the
// gather is branchless with clamped addresses; only delta writes are guarded.
// ---------------------------------------------------------------------------
__global__ __launch_bounds__(32) void codec_wavefront_kernel(
    const float* __restrict__ x,
    const float* __restrict__ W1, const float* __restrict__ b1,
    const float* __restrict__ W2, const float* __restrict__ b2,
    const float* __restrict__ W3, const float* __restrict__ b3,
    const float* __restrict__ W4, const float* __restrict__ b4,
    float* __restrict__ sum_out)
{
    // Weights pre-transposed to [out][in] so a B-fragment lane reads a
    // contiguous K-pair.  Out dims padded to tile multiples with zeros.
    __shared__ float sWt1[96 * 48];
    __shared__ float sWt2[32 * 96];
    __shared__ float sWt3[16 * 24];
    __shared__ float sWt4[16 * 12];
    __shared__ float sb1[96], sb2[32], sb3[16], sb4[16];
    // Rolling delta buffer: rows 0..2 = previous 3 interior rows (full width,
    // zero-padded cols 0..2 / 189..191), rows 3..18 = current 16-row group.
    __shared__ float sDelta[19 * IMG_W];
    // Inter-layer activations, widths padded so stores are unconditional.
    __shared__ float sH1[16 * 96];
    __shared__ float sH2[16 * 32];
    __shared__ float sH3[16 * 16];

    const int tid = threadIdx.x;
    const int ch  = blockIdx.x;                    // 0..23 == b*3 + c
    const float* xc = x + (size_t)ch * (IMG_H * IMG_W);

    // ---- preload / transpose weights into LDS --------------------------------
    for (int i = tid; i < 96 * 48; i += 32) { int n = i / 48, k = i % 48; sWt1[i] = W1[k * 96 + n]; }
    for (int i = tid; i < 32 * 96; i += 32) { int n = i / 96, k = i % 96; sWt2[i] = (n < 24) ? W2[k * 24 + n] : 0.f; }
    for (int i = tid; i < 16 * 24; i += 32) { int n = i / 24, k = i % 24; sWt3[i] = (n < 12) ? W3[k * 12 + n] : 0.f; }
    for (int i = tid; i < 16 * 12; i += 32) { int n = i / 12, k = i % 12; sWt4[i] = (n <  1) ? W4[k]        : 0.f; }
    for (int i = tid; i < 96; i += 32) sb1[i] = b1[i];
    for (int i = tid; i < 32; i += 32) sb2[i] = (i < 24) ? b2[i] : 0.f;
    if (tid < 16) { sb3[tid] = (tid < 12) ? b3[tid] : 0.f; sb4[tid] = (tid < 1) ? b4[0] : 0.f; }
    for (int i = tid; i < 19 * IMG_W; i += 32) sDelta[i] = 0.f;
    __syncthreads();

    const int m  = tid & 15;   // batch row this lane serves
    const int hi = tid >> 4;   // K-half selector for A/B fragments
    float accsum = 0.f;

    for (int g = 0; g < (INT_H + 15) / 16; ++g) {
        const int rows = (INT_H - g * 16 < 16) ? (INT_H - g * 16) : 16;
        const int tmax = (INT_W - 1) + 4 * (rows - 1);

        for (int t = 0; t <= tmax; ++t) {
            const int j   = t - 4 * m;                        // interior column
            const int jc  = iclamp(j, 0, INT_W - 1);          // safe column
            const int gic = iclamp(g * 16 + m, 0, INT_H - 1); // safe interior row

            // ---- build 16x48 feature matrix directly as A fragments ----------
            // k<24: image taps x[gi + k/7][j + k%7]  (k/7==3 -> same-row taps)
            // k>=24: delta taps sDelta[m + (k-24)/7][j + (k-24)%7]
            v2f A[12];
#pragma unroll
            for (int c = 0; c < 6; ++c) {
#pragma unroll
                for (int e = 0; e < 2; ++e) {
                    const int k = 4 * c + 2 * hi + e;
                    A[c][e] = xc[(gic + k / 7) * IMG_W + jc + k % 7];
                }
            }
#pragma unroll
            for (int c = 6; c < 12; ++c) {
#pragma unroll
                for (int e = 0; e < 2; ++e) {
                    const int kk = 4 * c + 2 * hi + e - 24;
                    A[c][e] = sDelta[(m + kk / 7) * IMG_W + jc + kk % 7];
                }
            }

            // ---- layer 1: 16x48 @ 48x96, 6 tiles x 12 K-chunks ---------------
#pragma unroll
            for (int tile = 0; tile < 6; ++tile) {
                v8f acc = {};
#pragma unroll
                for (int c = 0; c < 12; ++c) {
                    const int wb = (tile * 16 + m) * 48 + 4 * c + 2 * hi;
                    v2f B; B[0] = sWt1[wb]; B[1] = sWt1[wb + 1];
                    acc = __builtin_amdgcn_wmma_f32_16x16x4_f32(
                        false, A[c], false, B, (short)0, acc, false, false);
                }
                const float bias = sb1[tile * 16 + m];
#pragma unroll
                for (int v = 0; v < 8; ++v)
                    sH1[(v + 8 * hi) * 96 + tile * 16 + m] = TANH(acc[v] + bias);
            }
            __syncthreads();

            // ---- layer 2: 16x96 @ 96x24, 2 tiles x 24 K-chunks ---------------
#pragma unroll
            for (int tile = 0; tile < 2; ++tile) {
                v8f acc = {};
#pragma unroll
                for (int c = 0; c < 24; ++c) {
                    const int ab = m * 96 + 4 * c + 2 * hi;
                    v2f Af; Af[0] = sH1[ab]; Af[1] = sH1[ab + 1];
                    const int wb = (tile * 16 + m) * 96 + 4 * c + 2 * hi;
                    v2f B; B[0] = sWt2[wb]; B[1] = sWt2[wb + 1];
                    acc = __builtin_amdgcn_wmma_f32_16x16x4_f32(
                        false, Af, false, B, (short)0, acc, false, false);
                }
                const float bias = sb2[tile * 16 + m];
#pragma unroll
                for (int v = 0; v < 8; ++v)
                    sH2[(v + 8 * hi) * 32 + tile * 16 + m] = TANH(acc[v] + bias);
            }
            __syncthreads();

            // ---- layer 3: 16x24 @ 24x12, 1 tile x 6 K-chunks -----------------
            {
                v8f acc = {};
#pragma unroll
                for (int c = 0; c < 6; ++c) {
                    const int ab = m * 32 + 4 * c + 2 * hi;
                    v2f Af; Af[0] = sH2[ab]; Af[1] = sH2[ab + 1];
                    const int wb = m * 24 + 4 * c + 2 * hi;
                    v2f B; B[0] = sWt3[wb]; B[1] = sWt3[wb + 1];
                    acc = __builtin_amdgcn_wmma_f32_16x16x4_f32(
                        false, Af, false, B, (short)0, acc, false, false);
                }
                const float bias = sb3[m];
#pragma unroll
                for (int v = 0; v < 8; ++v)
                    sH3[(v + 8 * hi) * 16 + m] = TANH(acc[v] + bias);
            }
            __syncthreads();

            // ---- layer 4: 16x12 @ 12x1, 3 K-chunks; form deltas --------------
            {
                v8f acc = {};
#pragma unroll
                for (int c = 0; c < 3; ++c) {
                    const int ab = m * 16 + 4 * c + 2 * hi;
                    v2f Af; Af[0] = sH3[ab]; Af[1] = sH3[ab + 1];
                    const int wb = m * 12 + 4 * c + 2 * hi;
                    v2f B; B[0] = sWt4[wb]; B[1] = sWt4[wb + 1];
                    acc = __builtin_amdgcn_wmma_f32_16x16x4_f32(
                        false, Af, false, B, (short)0, acc, false, false);
                }
                if (m == 0) {                       // lanes 0 / 16 hold column 0
                    const float bias = sb4[0];
#pragma unroll
                    for (int v = 0; v < 8; ++v) {
                        const int row = v + 8 * hi;
                        const int jj  = t - 4 * row;
                        if (row < rows && jj >= 0 && jj < INT_W) {
                            const int gir = g * 16 + row;
                            const float pred = TANH(acc[v] + bias);
                            const float tgt  = xc[(gir + 3) * IMG_W + jj + 3];
                            const float d    = tgt - pred;
                            sDelta[(3 + row) * IMG_W + jj + 3] = d;
                            accsum += d * d;
                        }
                    }
                }
            }
            __syncthreads();
        } // wavefront steps

        // carry the last 3 computed rows into the prev-row slots
        for (int i = tid; i < 3 * IMG_W; i += 32) sDelta[i] = sDelta[16 * IMG_W + i];
        __syncthreads();
    }

    if (m == 0) atomicAdd(sum_out, accsum);   // lanes 0 and 16 hold partial sums
}

__global__ void codec_zero_kernel(float* ws) { ws[0] = 0.f; }

__global__ void codec_finalize_kernel(const float* ws, float* out) {
    // torch averages over (b, c, h-2, w) = 8*3*190*192 = 875520 elements
    out[0] = sqrtf(ws[0] / 875520.0f);
}

extern "C" void kernel_launch(void* const* d_in, const int* in_sizes, int n_in,
                              void* d_out, int out_size, void* d_ws, size_t ws_size,
                              hipStream_t stream) {
    (void)in_sizes; (void)n_in; (void)out_size; (void)ws_size;
    const float* x  = (const float*)d_in[0];
    const float* W1 = (const float*)d_in[1];
    const float* b1 = (const float*)d_in[2];
    const float* W2 = (const float*)d_in[3];
    const float* b2 = (const float*)d_in[4];
    const float* W3 = (const float*)d_in[5];
    const float* b3 = (const float*)d_in[6];
    const float* W4 = (const float*)d_in[7];
    const float* b4 = (const float*)d_in[8];
    float* out = (float*)d_out;
    float* ws  = (float*)d_ws;

    codec_zero_kernel<<<1, 1, 0, stream>>>(ws);
    codec_wavefront_kernel<<<NCH, 32, 0, stream>>>(x, W1, b1, W2, b2, W3, b3, W4, b4, ws);
    codec_finalize_kernel<<<1, 1, 0, stream>>>(ws, out);
}